// Modified_Hausdorff_Distance_Pytorch_35416300323048
// MI455X (gfx1250) — compile-verified
//
#include <hip/hip_runtime.h>
#include <math.h>

// Problem constants (reference: B=4, C=2, H=W=112, IGNORE_CLASS=0)
#define HWPIX 12544   // 112*112
#define IMW   112
#define NBATCH 4
#define ROWTILES_PER_WAVE 4          // 64 rows per wave
#define GRIDX (HWPIX / (16 * ROWTILES_PER_WAVE))   // 196
#define FLTMAX_BITS 0x7F7FFFFF       // bit pattern of FLT_MAX

typedef float v2f __attribute__((ext_vector_type(2)));
typedef float v8f __attribute__((ext_vector_type(8)));

// ---------------------------------------------------------------------------
// Workspace layout (bytes):
//   [0,32)    : int cntA[4], cntB[4]
//   [32,64)   : float sums[2][4]     (sums[0]=forward, sums[1]=reverse)
//   [64, ...) : float4 rowA[4*HWPIX], rowB[4*HWPIX],   (row form [x,y,sq,1])
//               float4 colA[4*HWPIX], colB[4*HWPIX]    (col form [-2x,-2y,1,sq])
// Pads: row pad [0,0,1e30,1]; col pad [0,0,1,1e30]  (d2 = sq_i + 1e30, never
// wins a min; always finite and non-negative, so float bits are int-monotone).
// ---------------------------------------------------------------------------

__global__ void k_init(unsigned* hdr, float4* enc, int nRowEnts, int nTot) {
  int t = blockIdx.x * blockDim.x + threadIdx.x;
  if (t < 16) hdr[t] = 0u;            // zero counters + sums (64 bytes)
  if (t < nTot) {
    float4 pad;
    if (t < nRowEnts) { pad.x = 0.f; pad.y = 0.f; pad.z = 1e30f; pad.w = 1.f; }
    else              { pad.x = 0.f; pad.y = 0.f; pad.z = 1.f;   pad.w = 1e30f; }
    enc[t] = pad;
  }
}

__global__ void k_compact(const float* __restrict__ pred,
                          const int* __restrict__ lab,
                          int* cntA, int* cntB,
                          float4* rowA, float4* rowB,
                          float4* colA, float4* colB) {
  int idx = blockIdx.x * blockDim.x + threadIdx.x;
  if (idx >= NBATCH * HWPIX) return;
  int b = idx / HWPIX;
  int p = idx - b * HWPIX;
  float p0 = pred[(b * 2 + 0) * HWPIX + p];
  float p1 = pred[(b * 2 + 1) * HWPIX + p];
  bool mA = p1 > p0;                 // argmax==1 (tie -> class 0, like jnp.argmax)
  bool mB = lab[b * HWPIX + p] == 1;
  float fi = (float)(p / IMW);       // meshgrid 'ij': first coord = row index
  float fj = (float)(p - (p / IMW) * IMW);
  float sq = fi * fi + fj * fj;
  float4 re; re.x = fi;         re.y = fj;         re.z = sq;  re.w = 1.f;
  float4 ce; ce.x = -2.f * fi;  ce.y = -2.f * fj;  ce.z = 1.f; ce.w = sq;
  if (mA) { int k = atomicAdd(&cntA[b], 1); rowA[b*HWPIX+k] = re; colA[b*HWPIX+k] = ce; }
  if (mB) { int k = atomicAdd(&cntB[b], 1); rowB[b*HWPIX+k] = re; colB[b*HWPIX+k] = ce; }
}

// One wave (32 threads) per 4 row tiles (64 rows).  dir=0: rows=A, cols=B
// (forward); dir=1: rows=B, cols=A (reverse).  Each B-tile load (8B/lane)
// feeds 4 V_WMMA_F32_16X16X4_F32 ops producing 4x(16x16) exact squared
// distances: row [x,y,sq,1] . col [-2x',-2y',1,sq'] = |p - p'|^2.
// Min-accumulation is done on the raw float bits with integer mins (all d2
// are non-negative finite) -> no v_max_num canonicalize ops in the hot loop.
__global__ void k_pair(const int* __restrict__ cntA, const int* __restrict__ cntB,
                       const float4* __restrict__ rowA, const float4* __restrict__ rowB,
                       const float4* __restrict__ colA, const float4* __restrict__ colB,
                       float* sums) {
  const int lane = threadIdx.x;
  const int blk  = blockIdx.x;
  const int b    = blockIdx.y;
  const int dir  = blockIdx.z;

  const int nRows = dir ? cntB[b] : cntA[b];
  const int nCols = dir ? cntA[b] : cntB[b];
  if (blk * (16 * ROWTILES_PER_WAVE) >= nRows) return;  // wave-uniform exit

  const float4* rowE = (dir ? rowB : rowA) + b * HWPIX;
  const float4* colE = (dir ? colA : colB) + b * HWPIX;

  // A fragments (16x4 f32): lane L holds row L&15; VGPR pair = K 0..1
  // (lanes 0-15) or K 2..3 (lanes 16-31).
  const int half = (lane < 16) ? 0 : 2;
  v2f a[ROWTILES_PER_WAVE];
  #pragma unroll
  for (int r = 0; r < ROWTILES_PER_WAVE; ++r) {
    int arow = blk * (16 * ROWTILES_PER_WAVE) + r * 16 + (lane & 15);
    a[r] = *(const v2f*)((const float*)(rowE + arow) + half);
  }

  int mn[ROWTILES_PER_WAVE][8];
  #pragma unroll
  for (int r = 0; r < ROWTILES_PER_WAVE; ++r)
    #pragma unroll
    for (int i = 0; i < 8; ++i) mn[r][i] = FLTMAX_BITS;

  const int nT = (nCols + 15) >> 4;
  for (int t = 0; t < nT; ++t) {
    // B fragment (4x16 f32): VGPR g, lanes 0-15 = K=g, lanes 16-31 = K=g+2,
    // column = lane&15.  Column form precomputed -> no per-iter transform.
    const int bcol = t * 16 + (lane & 15);
    v2f bb = *(const v2f*)((const float*)(colE + bcol) + half);
    #pragma unroll
    for (int r = 0; r < ROWTILES_PER_WAVE; ++r) {
      v8f c = {};
      v8f d = __builtin_amdgcn_wmma_f32_16x16x4_f32(
          /*neg_a=*/false, a[r], /*neg_b=*/false, bb,
          /*c_mod=*/(short)0, c, /*reuse_a=*/false, /*reuse_b=*/false);
      #pragma unroll
      for (int i = 0; i < 8; ++i) {
        int di = __float_as_int(d[i]);         // non-negative: int-monotone
        mn[r][i] = (di < mn[r][i]) ? di : mn[r][i];
      }
    }
  }

  // Min-reduce across the 16 columns (xor masks 1..8 stay within each half,
  // matching the C/D layout: VGPR i = row i / i+8, lane&15 = column).
  #pragma unroll
  for (int m = 1; m < 16; m <<= 1)
    #pragma unroll
    for (int r = 0; r < ROWTILES_PER_WAVE; ++r)
      #pragma unroll
      for (int i = 0; i < 8; ++i) {
        int o = __shfl_xor(mn[r][i], m, 32);
        mn[r][i] = (o < mn[r][i]) ? o : mn[r][i];
      }

  // Lane 0 holds rowmins for rows +0..7 of each tile, lane 16 for +8..15.
  if ((lane & 15) == 0) {
    float s = 0.f;
    #pragma unroll
    for (int r = 0; r < ROWTILES_PER_WAVE; ++r) {
      int base = (blk * ROWTILES_PER_WAVE + r) * 16 + ((lane >> 4) << 3);
      #pragma unroll
      for (int i = 0; i < 8; ++i)
        if (base + i < nRows) s += sqrtf(__int_as_float(mn[r][i]));
    }
    atomicAdd(&sums[dir * 4 + b], s);
  }
}

__global__ void k_final(const int* cntA, const int* cntB,
                        const float* sums, float* out) {
  if (threadIdx.x != 0 || blockIdx.x != 0) return;
  const float INFV = __builtin_inff();
  float fS = 0.f, rS = 0.f, mS = 0.f;
  for (int b = 0; b < NBATCH; ++b) {
    int na = cntA[b], nb = cntB[b];
    float f, r, m;
    if (na == 0) { f = INFV; r = INFV; m = INFV; }       // empty A -> all inf
    else {
      f = (nb == 0) ? INFV : sums[b] / (float)na;        // min over empty B = inf
      r = sums[4 + b] / (float)(nb > 0 ? nb : 1);        // nb==0 -> sum 0 -> 0
      m = fmaxf(f, r);
    }
    fS += f; rS += r; mS += m;
  }
  float F = fS * 0.25f, R = rS * 0.25f, M = mS * 0.25f;
  // finalize(): [class0=0, class1=v, mean(all)=v/2, mean(excl 0)=v]
  out[0] = 0.f; out[1] = M; out[2]  = M * 0.5f; out[3]  = M;  // MHD
  out[4] = 0.f; out[5] = F; out[6]  = F * 0.5f; out[7]  = F;  // FHD
  out[8] = 0.f; out[9] = R; out[10] = R * 0.5f; out[11] = R;  // RHD
}

extern "C" void kernel_launch(void* const* d_in, const int* in_sizes, int n_in,
                              void* d_out, int out_size, void* d_ws, size_t ws_size,
                              hipStream_t stream) {
  (void)in_sizes; (void)n_in; (void)out_size; (void)ws_size;
  const float* pred = (const float*)d_in[0];   // [4,2,112,112] f32
  const int*   lab  = (const int*)d_in[1];     // [4,112,112] i32
  float* out = (float*)d_out;                  // 12 floats (MHD|FHD|RHD)

  char* ws = (char*)d_ws;
  unsigned* hdr = (unsigned*)ws;
  int*   cntA = (int*)ws;
  int*   cntB = cntA + 4;
  float* sums = (float*)(ws + 32);
  float4* rowA = (float4*)(ws + 64);
  float4* rowB = rowA + NBATCH * HWPIX;
  float4* colA = rowB + NBATCH * HWPIX;
  float4* colB = colA + NBATCH * HWPIX;

  const int nRowEnts = 2 * NBATCH * HWPIX;     // row-form entries
  const int nTot     = 4 * NBATCH * HWPIX;     // row + col form entries
  k_init<<<(nTot + 255) / 256, 256, 0, stream>>>(hdr, rowA, nRowEnts, nTot);
  k_compact<<<(NBATCH * HWPIX + 255) / 256, 256, 0, stream>>>(
      pred, lab, cntA, cntB, rowA, rowB, colA, colB);
  k_pair<<<dim3(GRIDX, NBATCH, 2), 32, 0, stream>>>(
      cntA, cntB, rowA, rowB, colA, colB, sums);
  k_final<<<1, 32, 0, stream>>>(cntA, cntB, sums, out);
}